// FreqSparseInteractionBlock_51058571215405
// MI455X (gfx1250) — compile-verified
//
#include <hip/hip_runtime.h>
#include <math.h>

// ---------------------------------------------------------------------------
// FreqSparseInteractionBlock for MI455X (gfx1250, wave32, WMMA + TDM)
//   rfft(8192) -> transformer encoder (flash attention, f32 WMMA) ->
//   logits/sigmoid -> exact top-k(207) radix-select + sparse softmax ->
//   big einsum GEMM (TDM async LDS tiles + f32 WMMA) -> irfft ->
//   fused residual LayerNorm.
// Workspace requirement: ~250 MB (zero-initialized every call).
// ---------------------------------------------------------------------------

typedef __attribute__((ext_vector_type(2))) float v2f;
typedef __attribute__((ext_vector_type(8))) float v8f;
typedef __attribute__((ext_vector_type(4))) unsigned int u32x4;
typedef __attribute__((ext_vector_type(8))) int i32x8;
typedef __attribute__((ext_vector_type(4))) int i32x4;

#define B_BATCH 4
#define SEQ_L   8192
#define D_MODEL 512
#define D_MEM   256
#define NHEAD   8
#define HEAD_D  32
#define F_LEN   4097          // SEQ_L/2 + 1
#define F_PAD   4112          // 257 * 16
#define NEG_INF (-__builtin_inff())

#define EIN_KC  16            // K-chunk of the einsum GEMM (4112 = 257*16)
#define EIN_NT  128           // N columns per block in the einsum GEMM

#if __has_builtin(__builtin_amdgcn_tensor_load_to_lds) && \
    __has_builtin(__builtin_amdgcn_s_wait_tensorcnt)
#define HAVE_TDM 1
#else
#define HAVE_TDM 0
#endif

__device__ __forceinline__ v8f wmma4(v2f a, v2f b, v8f c) {
  // V_WMMA_F32_16X16X4_F32 : D = A(16x4) * B(4x16) + C(16x16)
  return __builtin_amdgcn_wmma_f32_16x16x4_f32(
      false, a, false, b, (short)0, c, false, false);
}

#if HAVE_TDM
// Issue one TDM 2D tile load Global->LDS (f32 elements, row-major tile).
// D# per CDNA5 ISA ch.8: group0 = {flags/count, lds_addr, global_addr, type},
// group1 = {data_size, tensor dims (= tile extents, tile always in-bounds),
// tile dims, dim0 row stride}. Groups 2/3 zero (2D tensor). Tracked by
// TENSORcnt; wave-uniform (caller must be a single wave, EXEC ignored).
// This toolchain exposes the 6-arg builtin:
//   (u32x4 g0, i32x8 g1, i32x4 g2, i32x4 g3, i32x8 g4, i32 cpol)
__device__ __forceinline__ void tdm_load_tile_f32(unsigned lds_byte_off,
                                                  const float* g,
                                                  unsigned tile_w,
                                                  unsigned tile_h,
                                                  unsigned row_stride_elems) {
  unsigned long long ga = (unsigned long long)(size_t)g;
  u32x4 g0;
  g0[0] = 1u;                                            // count=1, user mode
  g0[1] = lds_byte_off;                                  // lds_addr (bytes)
  g0[2] = (unsigned)ga;                                  // global_addr[31:0]
  g0[3] = (unsigned)((ga >> 32) & 0x01FFFFFFu) | (2u << 30);  // [56:32]|type=2
  i32x8 g1;
  g1[0] = 0x00020000;                                    // data_size=4B
  g1[1] = (int)((tile_w & 0xFFFFu) << 16);               // tensor_dim0 lo16
  g1[2] = (int)((tile_w >> 16) | ((tile_h & 0xFFFFu) << 16));  // d0 hi|d1 lo
  g1[3] = (int)((tile_h >> 16) | (tile_w << 16));        // d1 hi | tile_dim0
  g1[4] = (int)(tile_h & 0xFFFFu);                       // tile_dim1 (dim2=0)
  g1[5] = (int)row_stride_elems;                         // dim0_stride lo32
  g1[6] = 0;                                             // stride hi bits
  g1[7] = 0;
  i32x4 gz4 = {0, 0, 0, 0};
  i32x8 gz8 = {0, 0, 0, 0, 0, 0, 0, 0};
  __builtin_amdgcn_tensor_load_to_lds(g0, g1, gz4, gz4, gz8, 0);
}
#endif

// ============================== FFT kernels ================================
// One workgroup (256 threads) per (b, d) row: full 8192-pt complex FFT in LDS
// (radix-2 DIT, bit-reversed load, natural-order output).

__global__ void fft_fwd_kernel(const float* __restrict__ x,
                               float* __restrict__ Xf) {
  extern __shared__ float sm[];
  float* re = sm;
  float* im = sm + 8192;
  const int tid = threadIdx.x;
  const int b = blockIdx.x >> 9;
  const int d = blockIdx.x & 511;
  const size_t xbase = ((size_t)b * SEQ_L) * D_MODEL + d;

  for (int i = tid; i < 8192; i += 256) {
    int j = (int)(__brev((unsigned)i) >> 19);   // 13-bit reversal
    re[j] = x[xbase + (size_t)i * D_MODEL];
    im[j] = 0.f;
  }
  __syncthreads();

  for (int len = 2; len <= 8192; len <<= 1) {
    const int half = len >> 1;
    const float ang0 = -6.283185307179586f / (float)len;
    for (int bt = tid; bt < 4096; bt += 256) {
      int grp = bt / half;
      int pos = bt - grp * half;
      int i0 = grp * len + pos;
      int i1 = i0 + half;
      float s, c;
      __sincosf(ang0 * (float)pos, &s, &c);
      float xr = re[i1], xi = im[i1];
      float tr = xr * c - xi * s;
      float ti = xr * s + xi * c;
      float ur = re[i0], ui = im[i0];
      re[i1] = ur - tr; im[i1] = ui - ti;
      re[i0] = ur + tr; im[i0] = ui + ti;
    }
    __syncthreads();
  }

  float* Xre = Xf + ((size_t)(b * 2 + 0)) * F_PAD * D_MODEL;
  float* Xim = Xf + ((size_t)(b * 2 + 1)) * F_PAD * D_MODEL;
  for (int f = tid; f < F_LEN; f += 256) {
    Xre[(size_t)f * D_MODEL + d] = re[f];
    Xim[(size_t)f * D_MODEL + d] = im[f];
  }
}

__global__ void fft_inv_kernel(const float* __restrict__ Yf,
                               float* __restrict__ xout) {
  extern __shared__ float sm[];
  float* re = sm;
  float* im = sm + 8192;
  const int tid = threadIdx.x;
  const int b = blockIdx.x >> 9;
  const int d = blockIdx.x & 511;
  const float* Yre = Yf + ((size_t)(b * 2 + 0)) * F_PAD * D_MODEL;
  const float* Yim = Yf + ((size_t)(b * 2 + 1)) * F_PAD * D_MODEL;

  // Rebuild full Hermitian spectrum, bit-reversed positions.
  for (int i = tid; i < 8192; i += 256) {
    int j = (int)(__brev((unsigned)i) >> 19);
    int f = (i <= 4096) ? i : (8192 - i);
    float sgn = (i <= 4096) ? 1.f : -1.f;
    re[j] = Yre[(size_t)f * D_MODEL + d];
    im[j] = sgn * Yim[(size_t)f * D_MODEL + d];
  }
  __syncthreads();

  for (int len = 2; len <= 8192; len <<= 1) {
    const int half = len >> 1;
    const float ang0 = 6.283185307179586f / (float)len;   // inverse: + sign
    for (int bt = tid; bt < 4096; bt += 256) {
      int grp = bt / half;
      int pos = bt - grp * half;
      int i0 = grp * len + pos;
      int i1 = i0 + half;
      float s, c;
      __sincosf(ang0 * (float)pos, &s, &c);
      float xr = re[i1], xi = im[i1];
      float tr = xr * c - xi * s;
      float ti = xr * s + xi * c;
      float ur = re[i0], ui = im[i0];
      re[i1] = ur - tr; im[i1] = ui - ti;
      re[i0] = ur + tr; im[i0] = ui + ti;
    }
    __syncthreads();
  }

  const size_t obase = ((size_t)b * SEQ_L) * D_MODEL + d;
  const float invN = 1.f / 8192.f;
  for (int t = tid; t < 8192; t += 256)
    xout[obase + (size_t)t * D_MODEL] = re[t] * invN;
}

// ============================ WMMA GEMM kernels ============================
// Fragment layout (per ISA doc):
//   A(16x4):  m = lane&15, k = 2*(lane>>4)+{0,1}   (2 VGPRs / lane)
//   B(4x16):  n = lane&15, k = 2*(lane>>4)+{0,1}
//   C/D:      row = e + 8*(lane>>4), col = lane&15 (e = 0..7)

// Out[m][n] = act(scale * sum_k A[m][k] * W[n][k] + bias[n])   (B = W^T)
__global__ void gemm_wT_kernel(const float* __restrict__ A,
                               const float* __restrict__ W,
                               const float* __restrict__ bias,
                               float* __restrict__ Out,
                               int M, int N, int K, int lda, int ldw, int ldo,
                               float scale, int act) {
  const int wpb = blockDim.x >> 5;
  const int wave = blockIdx.x * wpb + (threadIdx.x >> 5);
  const int Mt = (M + 15) >> 4;
  const int Nt = (N + 15) >> 4;
  if (wave >= Mt * Nt) return;
  const int mt = wave / Nt, nt = wave % Nt;
  const int lane = threadIdx.x & 31;
  const int r = lane & 15, hi = lane >> 4;

  const float* Ar = A + (size_t)(mt * 16 + r) * lda + 2 * hi;
  const float* Wr = W + (size_t)(nt * 16 + r) * ldw + 2 * hi;
  v8f acc = {0.f, 0.f, 0.f, 0.f, 0.f, 0.f, 0.f, 0.f};
  for (int k = 0; k < K; k += 4) {
    v2f a, b;
    a.x = Ar[k]; a.y = Ar[k + 1];
    b.x = Wr[k]; b.y = Wr[k + 1];
    acc = wmma4(a, b, acc);
  }
  const int col = nt * 16 + r;
  const float bv = (bias != nullptr && col < N) ? bias[col] : 0.f;
#pragma unroll
  for (int e = 0; e < 8; ++e) {
    int row = mt * 16 + e + hi * 8;
    if (row < M && col < N) {
      float v = acc[e] * scale + bv;
      if (act == 1)  // exact GELU
        v = 0.5f * v * (1.f + erff(v * 0.70710678118654752f));
      Out[(size_t)row * ldo + col] = v;
    }
  }
}

// Einsum GEMM: Out[p][m][n] = sum_k A[m][k] * B[p][k][n]   (8 planes).
// Block = 256 threads (8 waves): one 16-row M tile x 128 N columns.
// K streamed in 16-wide chunks through double-buffered LDS tiles loaded by
// the Tensor Data Mover (wave 0 issues, TENSORcnt + barrier handshake);
// falls back to cooperative vector loads when the TDM builtin is absent.
__global__ void gemm_nn_tiled_kernel(const float* __restrict__ A,
                                     const float* __restrict__ Bm,
                                     float* __restrict__ Out,
                                     int M, int N, int K, int lda, int ldb,
                                     int ldo, size_t strideB, size_t strideO) {
  __shared__ float As[2][16 * EIN_KC];        // 16 rows x 16 k
  __shared__ float Bs[2][EIN_KC * EIN_NT];    // 16 k x 128 n
  const float* Bp = Bm + strideB * blockIdx.z;
  float* Op = Out + strideO * blockIdx.z;
  const int mt = blockIdx.x;                  // 16-row tile index
  const int nBase = blockIdx.y * EIN_NT;
  const int tid = threadIdx.x;
  const int wave = tid >> 5;                  // 0..7 -> 16 columns each
  const int lane = tid & 31;
  const int r = lane & 15, hi = lane >> 4;
  const int nOff = wave * 16;
  const int nChunks = K / EIN_KC;             // 257

  v8f acc = {0.f, 0.f, 0.f, 0.f, 0.f, 0.f, 0.f, 0.f};

#if HAVE_TDM
  const unsigned aOff0 = (unsigned)(size_t)&As[0][0];
  const unsigned aOff1 = (unsigned)(size_t)&As[1][0];
  const unsigned bOff0 = (unsigned)(size_t)&Bs[0][0];
  const unsigned bOff1 = (unsigned)(size_t)&Bs[1][0];
  const float* Arow = A + (size_t)(mt * 16) * lda;
  if (tid < 32) {   // wave-uniform: only wave 0 drives the TDM
    tdm_load_tile_f32(aOff0, Arow, EIN_KC, 16, (unsigned)lda);
    tdm_load_tile_f32(bOff0, Bp + nBase, EIN_NT, EIN_KC, (unsigned)ldb);
    __builtin_amdgcn_s_wait_tensorcnt((short)0);
  }
  __syncthreads();

  for (int c = 0; c < nChunks; ++c) {
    const int buf = c & 1;
    if (tid < 32 && c + 1 < nChunks) {        // prefetch next chunk
      const int k1 = (c + 1) * EIN_KC;
      tdm_load_tile_f32(buf ? aOff0 : aOff1, Arow + k1, EIN_KC, 16,
                        (unsigned)lda);
      tdm_load_tile_f32(buf ? bOff0 : bOff1, Bp + (size_t)k1 * ldb + nBase,
                        EIN_NT, EIN_KC, (unsigned)ldb);
    }
    const float* Ar = &As[buf][r * EIN_KC];
    const float* Br = &Bs[buf][nOff + r];
#pragma unroll
    for (int ks = 0; ks < EIN_KC / 4; ++ks) {
      v2f a, b;
      a.x = Ar[4 * ks + 2 * hi];
      a.y = Ar[4 * ks + 2 * hi + 1];
      b.x = Br[(4 * ks + 2 * hi) * EIN_NT];
      b.y = Br[(4 * ks + 2 * hi + 1) * EIN_NT];
      acc = wmma4(a, b, acc);
    }
    if (tid < 32 && c + 1 < nChunks)
      __builtin_amdgcn_s_wait_tensorcnt((short)0);
    __syncthreads();
  }
#else
  for (int c = 0; c < nChunks; ++c) {
    const int k0 = c * EIN_KC;
    As[0][tid] = A[(size_t)(mt * 16 + (tid >> 4)) * lda + k0 + (tid & 15)];
#pragma unroll
    for (int t = 0; t < (EIN_KC * EIN_NT) / 256; ++t) {
      int idx = tid + t * 256;
      Bs[0][idx] =
          Bp[(size_t)(k0 + idx / EIN_NT) * ldb + nBase + (idx & (EIN_NT - 1))];
    }
    __syncthreads();
    const float* Ar = &As[0][r * EIN_KC];
    const float* Br = &Bs[0][nOff + r];
#pragma unroll
    for (int ks = 0; ks < EIN_KC / 4; ++ks) {
      v2f a, b;
      a.x = Ar[4 * ks + 2 * hi];
      a.y = Ar[4 * ks + 2 * hi + 1];
      b.x = Br[(4 * ks + 2 * hi) * EIN_NT];
      b.y = Br[(4 * ks + 2 * hi + 1) * EIN_NT];
      acc = wmma4(a, b, acc);
    }
    __syncthreads();
  }
#endif

  const int col = nBase + nOff + r;
#pragma unroll
  for (int e = 0; e < 8; ++e) {
    int row = mt * 16 + e + hi * 8;
    if (row < M && col < N) Op[(size_t)row * ldo + col] = acc[e];
  }
}

// ========================= Flash attention kernel ==========================
// One wave per (16-query tile, head). Streams 257 key tiles with online
// softmax; QK^T and P*V both via f32 WMMA. qkv layout: [F_PAD][768]
// (q | k | v each 256 wide, head h at offset h*32).

__global__ void attn_kernel(const float* __restrict__ qkv,
                            float* __restrict__ ctxout, int F) {
  __shared__ float S[16][17];
  __shared__ float scl[16];
  __shared__ float ssum[16];
  const int qt = blockIdx.x;       // 0..256
  const int h  = blockIdx.y;       // 0..7
  const int lane = threadIdx.x;    // 0..31
  const int r = lane & 15, hi = lane >> 4;
  const float inv_sqrt_hd = 0.17677669529663687f;   // 1/sqrt(32)

  // Q tile A-fragments for all 8 k-chunks (K = HEAD_D = 32).
  const float* Qb = qkv + (size_t)(qt * 16 + r) * 768 + h * 32 + 2 * hi;
  v2f qa[8];
#pragma unroll
  for (int c = 0; c < 8; ++c) { qa[c].x = Qb[4 * c]; qa[c].y = Qb[4 * c + 1]; }

  v8f ctx0 = {0.f, 0.f, 0.f, 0.f, 0.f, 0.f, 0.f, 0.f};
  v8f ctx1 = {0.f, 0.f, 0.f, 0.f, 0.f, 0.f, 0.f, 0.f};
  float m = NEG_INF, ssf = 0.f;    // per-row stats, valid on lanes 0..15

  for (int kt = 0; kt < 257; ++kt) {
    // scores = Q * K^T (B[k=d][n=key] = K[key][d])
    const float* Kb = qkv + (size_t)(kt * 16 + r) * 768 + 256 + h * 32 + 2 * hi;
    v8f sc = {0.f, 0.f, 0.f, 0.f, 0.f, 0.f, 0.f, 0.f};
#pragma unroll
    for (int c = 0; c < 8; ++c) {
      v2f kb; kb.x = Kb[4 * c]; kb.y = Kb[4 * c + 1];
      sc = wmma4(qa[c], kb, sc);
    }
    const int gcol = kt * 16 + r;
#pragma unroll
    for (int e = 0; e < 8; ++e)
      S[e + 8 * hi][r] = (gcol < F) ? sc[e] * inv_sqrt_hd : NEG_INF;
    __syncthreads();

    if (lane < 16) {                // row = lane
      float tmax = NEG_INF;
#pragma unroll
      for (int c2 = 0; c2 < 16; ++c2) tmax = fmaxf(tmax, S[lane][c2]);
      float nm = fmaxf(m, tmax);
      float sclr = expf(m - nm);    // exp(-inf) = 0 on first tile
      float lsum = 0.f;
#pragma unroll
      for (int c2 = 0; c2 < 16; ++c2) {
        float p = expf(S[lane][c2] - nm);
        S[lane][c2] = p;            // S now holds P
        lsum += p;
      }
      ssf = ssf * sclr + lsum;
      m = nm;
      scl[lane] = sclr;
    }
    __syncthreads();

#pragma unroll
    for (int e = 0; e < 8; ++e) {
      float sc2 = scl[e + 8 * hi];
      ctx0[e] *= sc2; ctx1[e] *= sc2;
    }
    // ctx += P(16x16) * V(16x32) : 4 k-chunks, 2 column halves.
    const float* Vb = qkv + (size_t)(kt * 16) * 768 + 512 + h * 32;
#pragma unroll
    for (int c = 0; c < 4; ++c) {
      v2f pa; pa.x = S[r][4 * c + 2 * hi]; pa.y = S[r][4 * c + 2 * hi + 1];
      const float* Vr0 = Vb + (size_t)(4 * c + 2 * hi) * 768;
      const float* Vr1 = Vb + (size_t)(4 * c + 2 * hi + 1) * 768;
      v2f vb0, vb1;
      vb0.x = Vr0[r];      vb0.y = Vr1[r];
      vb1.x = Vr0[16 + r]; vb1.y = Vr1[16 + r];
      ctx0 = wmma4(pa, vb0, ctx0);
      ctx1 = wmma4(pa, vb1, ctx1);
    }
    __syncthreads();   // S reused next iteration
  }

  if (lane < 16) ssum[lane] = ssf;
  __syncthreads();
#pragma unroll
  for (int e = 0; e < 8; ++e) {
    int row = e + 8 * hi;
    int f = qt * 16 + row;
    if (f < F) {
      float inv = 1.f / ssum[row];
      ctxout[(size_t)f * D_MEM + h * 32 + r]      = ctx0[e] * inv;
      ctxout[(size_t)f * D_MEM + h * 32 + 16 + r] = ctx1[e] * inv;
    }
  }
}

// ======================= LayerNorm / pointwise kernels =====================

__global__ void add_ln_kernel(const float* __restrict__ A,
                              const float* __restrict__ Bm,
                              const float* __restrict__ w,
                              const float* __restrict__ bias,
                              float* __restrict__ Out) {
  __shared__ float red[256];
  const int row = blockIdx.x, tid = threadIdx.x;  // cols == 256 == blockDim
  const size_t base = (size_t)row * D_MEM;
  float v = A[base + tid] + Bm[base + tid];
  red[tid] = v; __syncthreads();
  for (int s = 128; s > 0; s >>= 1) { if (tid < s) red[tid] += red[tid + s]; __syncthreads(); }
  const float mu = red[0] * (1.f / D_MEM);
  __syncthreads();
  const float dv = v - mu;
  red[tid] = dv * dv; __syncthreads();
  for (int s = 128; s > 0; s >>= 1) { if (tid < s) red[tid] += red[tid + s]; __syncthreads(); }
  const float var = red[0] * (1.f / D_MEM);
  Out[base + tid] = dv * rsqrtf(var + 1e-5f) * w[tid] + bias[tid];
}

__global__ void final_ln_kernel(const float* __restrict__ x,
                                const float* __restrict__ xi,
                                float* __restrict__ Out) {
  __shared__ float red[256];
  const int row = blockIdx.x, tid = threadIdx.x;   // 512 cols, 2 per thread
  const size_t base = (size_t)row * D_MODEL;
  float v0 = x[base + tid]       + xi[base + tid];
  float v1 = x[base + 256 + tid] + xi[base + 256 + tid];
  red[tid] = v0 + v1; __syncthreads();
  for (int s = 128; s > 0; s >>= 1) { if (tid < s) red[tid] += red[tid + s]; __syncthreads(); }
  const float mu = red[0] * (1.f / D_MODEL);
  __syncthreads();
  float d0 = v0 - mu, d1 = v1 - mu;
  red[tid] = d0 * d0 + d1 * d1; __syncthreads();
  for (int s = 128; s > 0; s >>= 1) { if (tid < s) red[tid] += red[tid + s]; __syncthreads(); }
  const float is = rsqrtf(red[0] * (1.f / D_MODEL) + 1e-5f);
  Out[base + tid]       = d0 * is;
  Out[base + 256 + tid] = d1 * is;
}

__global__ void sigmoid_kernel(const float* __restrict__ L,
                               float* __restrict__ P, int F, int ldl) {
  size_t i = (size_t)blockIdx.x * blockDim.x + threadIdx.x;
  size_t n = (size_t)F * F;
  if (i >= n) return;
  int g = (int)(i / F), f = (int)(i % F);
  float v = L[(size_t)g * ldl + f];
  P[i] = 1.f / (1.f + expf(-v));
}

__global__ void pad_copy_kernel(const float* __restrict__ src,
                                float* __restrict__ dst, int rows, int cols,
                                int prows) {
  int i = blockIdx.x * blockDim.x + threadIdx.x;
  if (i >= prows * cols) return;
  int rr = i / cols;
  dst[i] = (rr < rows) ? src[i] : 0.f;
}

// ====================== top-k(207) + sparse softmax ========================
// One block per logits row: exact k-th-largest via 4-pass 8-bit radix select
// on order-preserving uint keys; ties taken in ascending index order (matches
// jax.lax.top_k). Row is overwritten in place with softmax weights (C).

__device__ __forceinline__ unsigned f2ord(float x) {
  unsigned b = __float_as_uint(x);
  return (b & 0x80000000u) ? ~b : (b | 0x80000000u);
}
__device__ __forceinline__ float ord2f(unsigned u) {
  unsigned b = (u & 0x80000000u) ? (u ^ 0x80000000u) : ~u;
  return __uint_as_float(b);
}

__global__ void topk_softmax_kernel(float* __restrict__ L, int F, int k) {
  __shared__ unsigned u[F_PAD];
  __shared__ unsigned hist[256];
  __shared__ unsigned sdata[256];
  __shared__ float fred[256];
  __shared__ unsigned bc[2];
  __shared__ unsigned tieBase;
  const int tid = threadIdx.x;
  float* row = L + (size_t)blockIdx.x * F_PAD;

  unsigned mymax = 0;
  for (int f = tid; f < F_PAD; f += 256) {
    unsigned b = 0;
    if (f < F) b = f2ord(row[f]);
    u[f] = b;
    if (b > mymax) mymax = b;
  }
  sdata[tid] = mymax; __syncthreads();
  for (int s = 128; s > 0; s >>= 1) {
    if (tid < s) { unsigned o = sdata[tid + s]; if (o > sdata[tid]) sdata[tid] = o; }
    __syncthreads();
  }
  const float rowmax = ord2f(sdata[0]);
  __syncthreads();

  unsigned prefix = 0;
  unsigned kk = (unsigned)k;
  for (int shift = 24; shift >= 0; shift -= 8) {
    hist[tid] = 0;
    __syncthreads();
    const unsigned hm = (shift == 24) ? 0u : (0xFFFFFFFFu << (shift + 8));
    for (int f = tid; f < F_PAD; f += 256) {
      unsigned v = u[f];
      if ((v & hm) == prefix) atomicAdd(&hist[(v >> shift) & 255u], 1u);
    }
    __syncthreads();
    if (tid == 0) {
      unsigned acc = 0; int b = 255;
      for (; b > 0; --b) {
        unsigned c = hist[b];
        if (acc + c >= kk) break;
        acc += c;
      }
      bc[0] = (unsigned)b; bc[1] = kk - acc;
    }
    __syncthreads();
    prefix |= bc[0] << shift;
    kk = bc[1];
    __syncthreads();
  }
  const unsigned tau = prefix;       // exact k-th largest key
  const unsigned tiesK = kk;         // #ties at tau to keep (index order)

  if (tid == 0) tieBase = 0;
  float lsum = 0.f;
  __syncthreads();
  for (int f0 = 0; f0 < F_PAD; f0 += 256) {
    const int f = f0 + tid;
    const unsigned v = (f < F_PAD) ? u[f] : 0u;
    const unsigned isTie = (f < F_PAD && v == tau) ? 1u : 0u;
    sdata[tid] = isTie;
    __syncthreads();
    for (int s = 1; s < 256; s <<= 1) {        // inclusive scan
      unsigned t = (tid >= s) ? sdata[tid - s] : 0u;
      __syncthreads();
      sdata[tid] += t;
      __syncthreads();
    }
    const unsigned incl = sdata[tid];
    const unsigned total = sdata[255];
    const unsigned myBase = tieBase;
    bool sel = false;
    if (f < F_PAD) {
      if (v > tau) sel = true;
      else if (isTie && (myBase + incl - 1u) < tiesK) sel = true;
    }
    float wv = 0.f;
    if (sel) { wv = expf(ord2f(v) - rowmax); lsum += wv; }
    if (f < F_PAD) u[f] = __float_as_uint(wv);
    __syncthreads();
    if (tid == 0) tieBase += total;
    __syncthreads();
  }
  fred[tid] = lsum; __syncthreads();
  for (int s = 128; s > 0; s >>= 1) { if (tid < s) fred[tid] += fred[tid + s]; __syncthreads(); }
  const float inv = 1.f / fred[0];
  for (int f = tid; f < F_PAD; f += 256)
    row[f] = __uint_as_float(u[f]) * inv;
}

// ================================ launcher =================================

static void launch_gemm_wT(const float* A, const float* W, const float* bias,
                           float* Out, int M, int N, int K, int lda, int ldw,
                           int ldo, float scale, int act, hipStream_t s) {
  int waves = ((M + 15) / 16) * ((N + 15) / 16);
  int grid = (waves + 3) / 4;
  gemm_wT_kernel<<<grid, 128, 0, s>>>(A, W, bias, Out, M, N, K, lda, ldw, ldo,
                                      scale, act);
}

extern "C" void kernel_launch(void* const* d_in, const int* in_sizes, int n_in,
                              void* d_out, int out_size, void* d_ws,
                              size_t ws_size, hipStream_t stream) {
  (void)in_sizes; (void)n_in; (void)out_size;
  const float* x_ts = (const float*)d_in[0];
  const float* M_frq = (const float*)d_in[1];
  const float* ipw = (const float*)d_in[2];
  const float* ipb = (const float*)d_in[3];
  const float* opw = (const float*)d_in[4];
  const float* opb = (const float*)d_in[5];
  const float* n1w = (const float*)d_in[6];
  const float* n1b = (const float*)d_in[7];
  const float* n2w = (const float*)d_in[8];
  const float* n2b = (const float*)d_in[9];
  const float* l1w = (const float*)d_in[10];
  const float* l1b = (const float*)d_in[11];
  const float* l2w = (const float*)d_in[12];
  const float* l2b = (const float*)d_in[13];
  float* out = (float*)d_out;

  float* ws = (float*)d_ws;
  const size_t S_X = (size_t)8 * F_PAD * D_MODEL;      // 8 planes (b x re/im)
  const size_t oX   = 0;                 // x_freq, later reused as x_ts_int
  const size_t oY   = oX + S_X;          // x_freq_int
  const size_t oL   = oY + S_X;          // logits -> C (ld F_PAD)
  const size_t oQKV = oL + (size_t)F_PAD * F_PAD;
  const size_t oCTX = oQKV + (size_t)F_PAD * 768;
  const size_t oAO  = oCTX + (size_t)F_PAD * D_MEM;
  const size_t oH   = oAO + (size_t)F_PAD * D_MEM;
  const size_t oT1  = oH + (size_t)F_PAD * D_MEM;
  const size_t oT2  = oT1 + (size_t)F_PAD * 1024;
  const size_t oMC  = oT2 + (size_t)F_PAD * D_MEM;
  const size_t oMP  = oMC + (size_t)F_PAD * D_MEM;
  const size_t totalF = oMP + (size_t)F_PAD * D_MEM;   // ~250 MB

  size_t zbytes = totalF * sizeof(float);
  if (zbytes > ws_size) zbytes = ws_size;
  (void)hipMemsetAsync(d_ws, 0, zbytes, stream);

  int top_k = (int)(25.0 * log((double)F_LEN));
  if (top_k < 1) top_k = 1;
  if (top_k > F_LEN) top_k = F_LEN;

  // 1) rfft: [4,8192,512] -> 8 planes [F_PAD][512]
  fft_fwd_kernel<<<B_BATCH * D_MODEL, 256, 2 * 8192 * sizeof(float), stream>>>(
      x_ts, ws + oX);

  // 2) transformer encoder layer on M_frq
  pad_copy_kernel<<<(F_PAD * D_MEM + 255) / 256, 256, 0, stream>>>(
      M_frq, ws + oMP, F_LEN, D_MEM, F_PAD);
  launch_gemm_wT(ws + oMP, ipw, ipb, ws + oQKV, F_LEN, 3 * D_MEM, D_MEM,
                 D_MEM, D_MEM, 768, 1.f, 0, stream);
  attn_kernel<<<dim3(F_PAD / 16, NHEAD), 32, 0, stream>>>(ws + oQKV, ws + oCTX,
                                                          F_LEN);
  launch_gemm_wT(ws + oCTX, opw, opb, ws + oAO, F_LEN, D_MEM, D_MEM, D_MEM,
                 D_MEM, D_MEM, 1.f, 0, stream);
  add_ln_kernel<<<F_LEN, 256, 0, stream>>>(ws + oMP, ws + oAO, n1w, n1b,
                                           ws + oH);
  launch_gemm_wT(ws + oH, l1w, l1b, ws + oT1, F_LEN, 4 * D_MEM, D_MEM, D_MEM,
                 D_MEM, 1024, 1.f, 1, stream);           // fused exact GELU
  launch_gemm_wT(ws + oT1, l2w, l2b, ws + oT2, F_LEN, D_MEM, 4 * D_MEM, 1024,
                 1024, D_MEM, 1.f, 0, stream);
  add_ln_kernel<<<F_LEN, 256, 0, stream>>>(ws + oH, ws + oT2, n2w, n2b,
                                           ws + oMC);

  // 3) logits = M_ctx @ M_ctx^T / 16 (padded region -> zeros)
  launch_gemm_wT(ws + oMC, ws + oMC, nullptr, ws + oL, F_PAD, F_PAD, D_MEM,
                 D_MEM, D_MEM, F_PAD, 1.f / 16.f, 0, stream);

  // 4) p_connect = sigmoid(logits) (before top-k overwrites logits)
  {
    size_t n = (size_t)F_LEN * F_LEN;
    sigmoid_kernel<<<(unsigned)((n + 255) / 256), 256, 0, stream>>>(
        ws + oL, out + (size_t)B_BATCH * SEQ_L * D_MODEL, F_LEN, F_PAD);
  }

  // 5) top-k + sparse softmax -> C (in place in logits buffer)
  topk_softmax_kernel<<<F_LEN, 256, 0, stream>>>(ws + oL, F_LEN, top_k);

  // 6) x_freq_int[p][g][d] = sum_f C[g][f] * x_freq[p][f][d], 8 planes,
  //    TDM-fed LDS tiles + f32 WMMA.
  gemm_nn_tiled_kernel<<<dim3(F_PAD / 16, D_MODEL / EIN_NT, 8), 256, 0,
                         stream>>>(ws + oL, ws + oX, ws + oY, F_LEN, D_MODEL,
                                   F_PAD, F_PAD, D_MODEL, D_MODEL,
                                   (size_t)F_PAD * D_MODEL,
                                   (size_t)F_PAD * D_MODEL);

  // 7) irfft -> x_ts_int (reuse x_freq region)
  fft_inv_kernel<<<B_BATCH * D_MODEL, 256, 2 * 8192 * sizeof(float), stream>>>(
      ws + oY, ws + oX);

  // 8) Z_out = LayerNorm(x_ts + x_ts_int), weight=1 bias=0
  final_ln_kernel<<<B_BATCH * SEQ_L, 256, 0, stream>>>(x_ts, ws + oX, out);
}